// ProposalLayer_21715354649161
// MI455X (gfx1250) — compile-verified
//
#include <hip/hip_runtime.h>
#include <hip/hip_bf16.h>
#include <stdint.h>

// Problem constants (fixed by the reference setup_inputs()).
#define B_DIM     16
#define N_DIM     16384
#define C_DIM     76
#define PRE_TOP   1024
#define POST_TOP  256
#define NMS_TH    0.85f
#define PI_F      3.14159265358979323846f   // rounds to 3.1415927f
#define TWO_PI_F  6.28318530717958647692f   // rounds to 6.2831855f

// ---------------------------------------------------------------------------
// CDNA5 async copy: global -> LDS, tracked by ASYNCcnt (gfx1250-only path).
// GVS addressing: mem = SADDR(64b SGPR pair) + VADDR(32b byte offset).
// vdst holds the LDS byte address (low 32 bits of a generic LDS pointer).
// ---------------------------------------------------------------------------
__device__ __forceinline__ void async_ld_b32(uint32_t lds_off, uint32_t voff,
                                             const void* sbase) {
  asm volatile("global_load_async_to_lds_b32 %0, %1, %2"
               :: "v"(lds_off), "v"(voff), "s"(sbase)
               : "memory");
}
__device__ __forceinline__ void wait_async0() {
  asm volatile("s_wait_asynccnt 0" ::: "memory");
}

// ---------------------------------------------------------------------------
// Force-inlined helpers (no lambdas: avoids the un-inlined closure + flat
// loads of captured state seen in the previous build's disassembly).
// ---------------------------------------------------------------------------

// Async-stage one 1024-float chunk into LDS, pack stable-sort keys.
// key = (order-flipped f32 bits << 32) | ~index  -> descending u64 order
//       == stable jnp.argsort(-scores) (ties: smaller index first).
__device__ __forceinline__ void load_chunk(unsigned long long* keys,
                                           float* stage, uint32_t stageBase,
                                           const float* sb, int c, int dstBase,
                                           int tid) {
  async_ld_b32(stageBase + (uint32_t)tid * 4u,
               (uint32_t)((c * PRE_TOP + tid) * 4), (const void*)sb);
  wait_async0();
  __syncthreads();
  float v = stage[tid];
  uint32_t bits = __float_as_uint(v);
  uint32_t f = bits ^ ((bits & 0x80000000u) ? 0xFFFFFFFFu : 0x80000000u);
  uint32_t idx = (uint32_t)(c * PRE_TOP + tid);
  keys[dstBase + tid] =
      ((unsigned long long)f << 32) | (unsigned long long)(~idx);
  __syncthreads();
}

// In-LDS bitonic sort of keys[base..base+1023], descending. 1024 threads.
__device__ __forceinline__ void bitonic_sort_desc_1024(unsigned long long* keys,
                                                       int base, int tid) {
  for (int k = 2; k <= PRE_TOP; k <<= 1) {
    for (int j = k >> 1; j > 0; j >>= 1) {
      int ixj = tid ^ j;
      if (ixj > tid) {
        unsigned long long a  = keys[base + tid];
        unsigned long long c2 = keys[base + ixj];
        bool desc = ((tid & k) == 0);
        if ((a < c2) == desc) { keys[base + tid] = c2; keys[base + ixj] = a; }
      }
      __syncthreads();
    }
  }
}

// Descending bitonic merge of a bitonic 1024-sequence in keys[0..1023].
__device__ __forceinline__ void bitonic_merge_desc_1024(unsigned long long* keys,
                                                        int tid) {
  for (int j = PRE_TOP >> 1; j > 0; j >>= 1) {
    int ixj = tid ^ j;
    if (ixj > tid) {
      unsigned long long x = keys[tid], y = keys[ixj];
      if (x < y) { keys[tid] = y; keys[ixj] = x; }
    }
    __syncthreads();
  }
}

// ---------------------------------------------------------------------------
// Kernel 1: per-batch stable top-1024 of 16384 scores, descending.
// Iterative bitonic top-k: keep sorted top-1024, fold in 1024-chunks via
// elementwise-max merge C[t] = max(A[t], B[1023-t]) (bitonic), then clean up.
// ---------------------------------------------------------------------------
__global__ __launch_bounds__(1024) void topk1024_kernel(
    const float* __restrict__ scores,
    int*   __restrict__ topIdx,
    float* __restrict__ topScore) {
  const int b   = blockIdx.x;
  const int tid = threadIdx.x;

  __shared__ unsigned long long keys[2 * PRE_TOP];   // 16 KB
  __shared__ float stage[PRE_TOP];                   // 4 KB async staging

  const float* sb = scores + (size_t)b * N_DIM;
  const uint32_t stageBase = (uint32_t)(size_t)(&stage[0]);  // LDS byte offset

  load_chunk(keys, stage, stageBase, sb, 0, 0, tid);
  bitonic_sort_desc_1024(keys, 0, tid);

  for (int c = 1; c < N_DIM / PRE_TOP; ++c) {
    load_chunk(keys, stage, stageBase, sb, c, PRE_TOP, tid);
    bitonic_sort_desc_1024(keys, PRE_TOP, tid);
    // Top-1024 of (A desc ++ B desc): thread t is sole reader/writer of keys[t].
    unsigned long long a  = keys[tid];
    unsigned long long bb = keys[2 * PRE_TOP - 1 - tid];
    keys[tid] = (a > bb) ? a : bb;
    __syncthreads();
    bitonic_merge_desc_1024(keys, tid);
  }

  unsigned long long kk = keys[tid];
  int idx = (int)(~((uint32_t)kk));
  topIdx[b * PRE_TOP + tid]   = idx;
  topScore[b * PRE_TOP + tid] = sb[idx];
}

// ---------------------------------------------------------------------------
// Kernel 2: decode only the selected 1024 rows per batch, then greedy BEV-NMS
// with exact reference semantics, then compact to <=256 outputs (zero-padded).
// One block (256 threads = 8 wave32) per batch.
// ---------------------------------------------------------------------------
__global__ __launch_bounds__(256) void decode_nms_kernel(
    const float* __restrict__ reg,       // (B,N,76)
    const float* __restrict__ xyz,       // (B,N,3)
    const float* __restrict__ meanSize,  // (3,)
    const int*   __restrict__ topIdx,    // (B,1024)
    const float* __restrict__ topScore,  // (B,1024)
    float* __restrict__ outBoxes,        // (B,256,7)
    float* __restrict__ outScores) {     // (B,256)
  const int b = blockIdx.x, tid = threadIdx.x;

  __shared__ float prS[PRE_TOP * 7];                       // 28 KB
  __shared__ float scS[PRE_TOP];                           // 4 KB
  __shared__ float bx1[PRE_TOP], bz1[PRE_TOP];
  __shared__ float bx2[PRE_TOP], bz2[PRE_TOP];             // 16 KB
  __shared__ float areaS[PRE_TOP];                         // 4 KB
  __shared__ int   keepS[PRE_TOP];                         // 4 KB
  __shared__ int   posS[PRE_TOP];                          // 4 KB
  __shared__ int   selR[POST_TOP];                         // 1 KB
  __shared__ int   nKept;

  const float ms0 = meanSize[0], ms1 = meanSize[1], ms2 = meanSize[2];
  const float apc = TWO_PI_F / 12.0f;

  // ---- decode the 1024 selected proposals (gathered 304B rows) ----
  for (int r = tid; r < PRE_TOP; r += 256) {
    int gi = topIdx[b * PRE_TOP + r];
    const float* rg   = reg + ((size_t)b * N_DIM + (size_t)gi) * C_DIM;
    const float* pxyz = xyz + ((size_t)b * N_DIM + (size_t)gi) * 3;

    // First-occurrence argmax over 12 bins (x, z, heading).
    int xb = 0, zb = 0, rb = 0;
    { float best = rg[0];  for (int k = 1; k < 12; ++k) { float v = rg[k];      if (v > best) { best = v; xb = k; } } }
    { float best = rg[12]; for (int k = 1; k < 12; ++k) { float v = rg[12 + k]; if (v > best) { best = v; zb = k; } } }
    { float best = rg[49]; for (int k = 1; k < 12; ++k) { float v = rg[49 + k]; if (v > best) { best = v; rb = k; } } }

    float px = (float)xb * 0.5f + 0.25f - 3.0f + rg[24 + xb] * 0.5f;
    float pz = (float)zb * 0.5f + 0.25f - 3.0f + rg[36 + zb] * 0.5f;
    float py = pxyz[1] + rg[48];

    float rres = rg[61 + rb];
    float ry = fmodf((float)rb * apc + rres * (apc * 0.5f), TWO_PI_F);
    if (ry < 0.0f)  ry += TWO_PI_F;     // jnp mod: sign of divisor
    if (ry > PI_F)  ry -= TWO_PI_F;

    float h = rg[73] * ms0 + ms0;
    float w = rg[74] * ms1 + ms1;
    float l = rg[75] * ms2 + ms2;
    px += pxyz[0];
    pz += pxyz[2];
    py += 0.5f * h;                      // props[:,1] += props[:,3]/2

    prS[r * 7 + 0] = px; prS[r * 7 + 1] = py; prS[r * 7 + 2] = pz;
    prS[r * 7 + 3] = h;  prS[r * 7 + 4] = w;  prS[r * 7 + 5] = l;
    prS[r * 7 + 6] = ry;
    scS[r] = topScore[b * PRE_TOP + r];

    float hl = 0.5f * l, hw = 0.5f * w;
    float X1 = px - hl, Z1 = pz - hw, X2 = px + hl, Z2 = pz + hw;
    bx1[r] = X1; bz1[r] = Z1; bx2[r] = X2; bz2[r] = Z2;
    areaS[r] = (X2 - X1) * (Z2 - Z1);    // match reference FP order
    keepS[r] = 1;
  }
  __syncthreads();

  // ---- greedy sequential NMS (suppressor must still be kept at step i) ----
  for (int i = 0; i < PRE_TOP; ++i) {
    if (keepS[i]) {
      float ax1 = bx1[i], az1 = bz1[i], ax2 = bx2[i], az2 = bz2[i];
      float aa = areaS[i];
      for (int r = tid; r < PRE_TOP; r += 256) {
        if (r > i && keepS[r]) {
          float lx = fmaxf(ax1, bx1[r]);
          float lz = fmaxf(az1, bz1[r]);
          float rx = fminf(ax2, bx2[r]);
          float rz = fminf(az2, bz2[r]);
          float iw = fmaxf(rx - lx, 0.0f);
          float ih = fmaxf(rz - lz, 0.0f);
          float inter = iw * ih;
          float iou = inter / (aa + areaS[r] - inter + 1e-8f);
          if (iou > NMS_TH) keepS[r] = 0;
        }
      }
    }
    __syncthreads();
  }

  // ---- rank-order compaction ----
  if (tid == 0) {
    int c = 0;
    for (int r = 0; r < PRE_TOP; ++r) { posS[r] = c; c += keepS[r]; }
    nKept = c;
  }
  __syncthreads();
  for (int r = tid; r < PRE_TOP; r += 256) {
    if (keepS[r]) { int p = posS[r]; if (p < POST_TOP) selR[p] = r; }
  }
  __syncthreads();

  // ---- one writer per output slot: kept box or zero (d_out is poisoned) ----
  float* ob = outBoxes  + (size_t)b * POST_TOP * 7;
  float* os = outScores + (size_t)b * POST_TOP;
  {
    int p = tid;  // 256 threads == POST_TOP slots
    if (p < nKept) {
      int r = selR[p];
#pragma unroll
      for (int cc = 0; cc < 7; ++cc) ob[p * 7 + cc] = prS[r * 7 + cc];
      os[p] = scS[r];
    } else {
#pragma unroll
      for (int cc = 0; cc < 7; ++cc) ob[p * 7 + cc] = 0.0f;
      os[p] = 0.0f;
    }
  }
}

// ---------------------------------------------------------------------------
// Host launcher. d_in order: rpn_scores, rpn_reg, xyz, mean_size (all f32).
// d_out: [16*256*7 boxes | 16*256 scores]. d_ws: topIdx (64KB) + topScore (64KB).
// ---------------------------------------------------------------------------
extern "C" void kernel_launch(void* const* d_in, const int* in_sizes, int n_in,
                              void* d_out, int out_size, void* d_ws, size_t ws_size,
                              hipStream_t stream) {
  (void)in_sizes; (void)n_in; (void)out_size; (void)ws_size;
  const float* scores = (const float*)d_in[0];
  const float* reg    = (const float*)d_in[1];
  const float* xyz    = (const float*)d_in[2];
  const float* msz    = (const float*)d_in[3];

  float* outBoxes  = (float*)d_out;
  float* outScores = (float*)d_out + (size_t)B_DIM * POST_TOP * 7;

  int*   topIdx   = (int*)d_ws;
  float* topScore = (float*)((char*)d_ws + (size_t)B_DIM * PRE_TOP * sizeof(int));

  topk1024_kernel<<<B_DIM, 1024, 0, stream>>>(scores, topIdx, topScore);
  decode_nms_kernel<<<B_DIM, 256, 0, stream>>>(reg, xyz, msz, topIdx, topScore,
                                               outBoxes, outScores);
}